// PPNet_69166153335416
// MI455X (gfx1250) — compile-verified
//
#include <hip/hip_runtime.h>
#include <math.h>

#define NN      50000     // nodes (== 3125 * 16, exact tiling)
#define NE      600000    // edges
#define NG      256       // graphs
#define HID     128
#define NLAYERS 4
#define NPROT   20        // 10 pos + 10 neg
#define EPSF    1e-12f

typedef __attribute__((ext_vector_type(2))) float v2f;
typedef __attribute__((ext_vector_type(8))) float v8f;

// Relaxed, agent-scope, non-returning f32 add -> native global_atomic_add_f32
__device__ __forceinline__ void fadd_relaxed(float* p, float v) {
    (void)__hip_atomic_fetch_add(p, v, __ATOMIC_RELAXED, __HIP_MEMORY_SCOPE_AGENT);
}

// ---------------------------------------------------------------- utilities
__global__ void k_zero4(float4* __restrict__ p, int n4) {
    const float4 z = {0.f, 0.f, 0.f, 0.f};
    for (int i = blockIdx.x * blockDim.x + threadIdx.x; i < n4; i += gridDim.x * blockDim.x)
        p[i] = z;
}

__global__ void k_deg(const int* __restrict__ dst, float* __restrict__ deg, int ne) {
    int i = blockIdx.x * blockDim.x + threadIdx.x;
    if (i < ne) fadd_relaxed(&deg[dst[i]], 1.0f);
}

__global__ void k_invdeg(float* __restrict__ deg, int n) {
    int i = blockIdx.x * blockDim.x + threadIdx.x;
    if (i < n) deg[i] = 1.0f / fmaxf(deg[i], 1.0f);
}

// ------------------------------------------------- embedding GEMM (fp32 WMMA)
// x[NN,128] = h[NN,128] @ W^T[128,128] + b ; one 16x16 tile per wave,
// 8 waves/block cover n0 = 0..127, blockIdx.x picks the 16-row stripe.
__global__ void k_emb(const float* __restrict__ h, const float* __restrict__ W,
                      const float* __restrict__ b, float* __restrict__ x) {
    const int wave = threadIdx.x >> 5;
    const int lane = threadIdx.x & 31;
    const int m0   = blockIdx.x * 16;
    const int n0   = wave * 16;
    const int half = lane >> 4;     // K-half selector
    const int sub  = lane & 15;     // M for A-frag, N for B/C-frag
    const int arow = m0 + sub;
    const int bcol = n0 + sub;      // B[k][n] = W[n*128 + k]
    v8f acc = {};
    for (int k0 = 0; k0 < HID; k0 += 4) {
        const int ka = k0 + 2 * half;        // VGPR v holds K = ka + v
        v2f a, bf;
        a.x  = h[arow * HID + ka];
        a.y  = h[arow * HID + ka + 1];
        bf.x = W[bcol * HID + ka];
        bf.y = W[bcol * HID + ka + 1];
        acc = __builtin_amdgcn_wmma_f32_16x16x4_f32(false, a, false, bf,
                                                    (short)0, acc, false, false);
    }
    const float bias = b[n0 + sub];
#pragma unroll
    for (int v = 0; v < 8; ++v) {
        const int m = m0 + v + 8 * half;     // C/D layout: M = v + 8*half
        x[m * HID + n0 + sub] = acc[v] + bias;
    }
}

// ------------------------------------------------------- edge scatter (atomics)
// neigh[dst] += x[src]; one wave per edge, float4 per lane (32*4 = 128 floats)
__global__ void k_agg(const float* __restrict__ x, const int* __restrict__ src,
                      const int* __restrict__ dst, float* __restrict__ neigh, int ne) {
    const int e = blockIdx.x * 8 + (threadIdx.x >> 5);
    if (e >= ne) return;
    const int lane = threadIdx.x & 31;
    const int s = src[e], d = dst[e];
    const float4 v = ((const float4*)(x + (size_t)s * HID))[lane];
    float* nd = neigh + (size_t)d * HID + lane * 4;
    fadd_relaxed(nd + 0, v.x);
    fadd_relaxed(nd + 1, v.y);
    fadd_relaxed(nd + 2, v.z);
    fadd_relaxed(nd + 3, v.w);
}

// --------------------------------- fused layer: GEMM(K=256) + norm + relu + res
// x <- x + relu(normalize(concat(x, neigh*inv_deg) @ Wl^T + bl))   (in place;
// each block exclusively owns rows m0..m0+15 for read and write)
__global__ void k_layer(float* __restrict__ x, const float* __restrict__ neigh,
                        const float* __restrict__ invdeg,
                        const float* __restrict__ Wl, const float* __restrict__ bl) {
    __shared__ float tile[16][HID + 4];
    __shared__ float scale[16];
    const int wave = threadIdx.x >> 5;
    const int lane = threadIdx.x & 31;
    const int m0   = blockIdx.x * 16;
    const int n0   = wave * 16;
    const int half = lane >> 4;
    const int sub  = lane & 15;
    const int arow = m0 + sub;
    const float idg = invdeg[arow];
    v8f acc = {};
    for (int k0 = 0; k0 < 2 * HID; k0 += 4) {
        const int ka = k0 + 2 * half;
        v2f a, bf;
        if (ka < HID) {                     // first half of concat: x
            a.x = x[arow * HID + ka];
            a.y = x[arow * HID + ka + 1];
        } else {                            // second half: mean-aggregated neigh
            a.x = neigh[arow * HID + ka - HID] * idg;
            a.y = neigh[arow * HID + ka - HID + 1] * idg;
        }
        bf.x = Wl[(n0 + sub) * (2 * HID) + ka];
        bf.y = Wl[(n0 + sub) * (2 * HID) + ka + 1];
        acc = __builtin_amdgcn_wmma_f32_16x16x4_f32(false, a, false, bf,
                                                    (short)0, acc, false, false);
    }
    const float bias = bl[n0 + sub];
#pragma unroll
    for (int v = 0; v < 8; ++v)
        tile[v + 8 * half][n0 + sub] = acc[v] + bias;
    __syncthreads();
    if (threadIdx.x < 16) {                 // per-row L2 norm
        float ss = 0.0f;
        for (int c = 0; c < HID; ++c) { float t = tile[threadIdx.x][c]; ss += t * t; }
        scale[threadIdx.x] = 1.0f / fmaxf(sqrtf(ss), EPSF);
    }
    __syncthreads();
    for (int i = threadIdx.x; i < 16 * HID; i += 256) {
        const int r = i >> 7, c = i & (HID - 1);
        const float v = fmaxf(tile[r][c] * scale[r], 0.0f);
        const size_t g = (size_t)(m0 + r) * HID + c;
        x[g] = x[g] + v;                    // residual, in place
    }
}

// --------------------------------------------------------------- graph pooling
__global__ void k_pool(const float* __restrict__ x, const int* __restrict__ gid,
                       float* __restrict__ hgsum, float* __restrict__ cnt, int nn) {
    const int node = blockIdx.x * 8 + (threadIdx.x >> 5);
    if (node >= nn) return;
    const int lane = threadIdx.x & 31;
    const int g = gid[node];
    const float4 v = ((const float4*)(x + (size_t)node * HID))[lane];
    float* hs = hgsum + (size_t)g * HID + lane * 4;
    fadd_relaxed(hs + 0, v.x);
    fadd_relaxed(hs + 1, v.y);
    fadd_relaxed(hs + 2, v.z);
    fadd_relaxed(hs + 3, v.w);
    if (lane == 0) fadd_relaxed(&cnt[g], 1.0f);
}

// -------------------------------------------- prototype head (one wave / graph)
__global__ void k_head(const float* __restrict__ hgsum, const float* __restrict__ cnt,
                       const float* __restrict__ ppos, const float* __restrict__ pneg,
                       const float* __restrict__ wfc, float* __restrict__ out) {
    const int g = blockIdx.x;
    const int lane = threadIdx.x;           // 32 threads
    const float ic = 1.0f / fmaxf(cnt[g], 1.0f);
    float hg[4];
#pragma unroll
    for (int j = 0; j < 4; ++j) hg[j] = hgsum[(size_t)g * HID + lane + 32 * j] * ic;
    float y = 0.0f;
    for (int p = 0; p < NPROT; ++p) {
        const float* pr = (p < NPROT / 2) ? (ppos + (size_t)p * HID)
                                          : (pneg + (size_t)(p - NPROT / 2) * HID);
        float d2 = 0.0f;
#pragma unroll
        for (int j = 0; j < 4; ++j) {
            const float df = hg[j] - pr[lane + 32 * j];
            d2 += df * df;
        }
        for (int o = 16; o > 0; o >>= 1) d2 += __shfl_xor(d2, o, 32);
        y += logf((d2 + 1.0f) / (d2 + EPSF)) * wfc[p];
    }
    if (lane == 0) out[g] = 1.0f / (1.0f + expf(-y));
}

// ------------------------------------------------------------------ launcher
extern "C" void kernel_launch(void* const* d_in, const int* in_sizes, int n_in,
                              void* d_out, int out_size, void* d_ws, size_t ws_size,
                              hipStream_t stream) {
    const float* h        = (const float*)d_in[0];
    const int*   src      = (const int*)  d_in[1];
    const int*   dst      = (const int*)  d_in[2];
    const int*   gids     = (const int*)  d_in[3];
    const float* W_emb    = (const float*)d_in[4];
    const float* b_emb    = (const float*)d_in[5];
    const float* W_layers = (const float*)d_in[6];
    const float* b_layers = (const float*)d_in[7];
    const float* p_pos    = (const float*)d_in[8];
    const float* p_neg    = (const float*)d_in[9];
    const float* W_fc     = (const float*)d_in[10];
    float*       out      = (float*)d_out;

    // workspace carve-up (all fp32)
    float* x     = (float*)d_ws;            // NN*HID
    float* neigh = x + (size_t)NN * HID;    // NN*HID
    float* ideg  = neigh + (size_t)NN * HID;// NN   (deg -> inv_deg in place)
    float* hgsum = ideg + NN;               // NG*HID
    float* cnt   = hgsum + (size_t)NG * HID;// NG

    const int ZB = 1024;                    // blocks for grid-stride zero fills

    // degree / inverse degree
    k_zero4<<<64, 256, 0, stream>>>((float4*)ideg, NN / 4);
    k_deg<<<(NE + 255) / 256, 256, 0, stream>>>(dst, ideg, NE);
    k_invdeg<<<(NN + 255) / 256, 256, 0, stream>>>(ideg, NN);

    // embedding GEMM (fp32 WMMA)
    k_emb<<<NN / 16, 256, 0, stream>>>(h, W_emb, b_emb, x);

    // GraphSage layers
    for (int l = 0; l < NLAYERS; ++l) {
        k_zero4<<<ZB, 256, 0, stream>>>((float4*)neigh, NN * HID / 4);
        k_agg<<<(NE + 7) / 8, 256, 0, stream>>>(x, src, dst, neigh, NE);
        k_layer<<<NN / 16, 256, 0, stream>>>(x, neigh, ideg,
                                             W_layers + (size_t)l * HID * 2 * HID,
                                             b_layers + (size_t)l * HID);
    }

    // per-graph mean pool + prototype head
    k_zero4<<<32, 256, 0, stream>>>((float4*)hgsum, NG * HID / 4);
    k_zero4<<<1, 64, 0, stream>>>((float4*)cnt, NG / 4);
    k_pool<<<(NN + 7) / 8, 256, 0, stream>>>(x, gids, hgsum, cnt, NN);
    k_head<<<NG, 32, 0, stream>>>(hgsum, cnt, p_pos, p_neg, W_fc, out);
}